// ShapleyQMixer_85289460564474
// MI455X (gfx1250) — compile-verified
//
#include <hip/hip_runtime.h>
#include <hip/hip_bf16.h>

// ---------------------------------------------------------------------------
// ShapleyQMixer on MI455X (gfx1250, wave32).
// Shapes: B=32 T=64 N=16 A=20 D=128 S=32 E=64 H=256, batch = 2048.
//
// Pipeline:
//   1..6) fp32 WMMA GEMMs (V_WMMA_F32_16X16X4_F32) for the hypernetwork,
//         computed once per batch element (st_rep rows repeat N=16x).
//         Each wave owns a 16x64 C strip (4 accumulators) so one A fragment
//         feeds 4 WMMAs -> ~2.5x fewer VMEM issues per matrix op; B rows are
//         prefetched 2 K-steps ahead (global_prefetch_b8).
//   7)    coalition kernel: exclusive prefix sums of permuted actions,
//         gathered at g=gc[b,s,i]; cnt==g since gc rows are permutations.
//   8)    fused final kernel: hidden = elu(inputs@w1 + b1),
//         y = <hidden, w_final> + v, w_est = |y|, q_tot = <w_est, agent_qs>.
// ---------------------------------------------------------------------------

typedef __attribute__((ext_vector_type(2))) float v2f;
typedef __attribute__((ext_vector_type(8))) float v8f;

#define BATCH 2048
#define NAG   16
#define NACT  20
#define NS    32
#define NE    64

// ---------------------------------------------------------------------------
// fp32 WMMA GEMM: C[M,N] = act(A[M,K] @ W[K,N] + bias[N]),  N % 64 == 0.
// ACT: 0 = identity, 1 = relu, 2 = abs.
// One wave per 16(M) x 64(N) strip = 4 C tiles; K stepped by 4 per
// V_WMMA_F32_16X16X4_F32.
// A-frag (16x4, 2 VGPR): M = lane%16; lanes 0-15 hold K={k,k+1},
//   lanes 16-31 hold K={k+2,k+3}.
// B-frag (4x16, 2 VGPR): N = lane%16; lanes 0-15 hold K={k,k+1},
//   lanes 16-31 hold K={k+2,k+3}.
// C/D (16x16, 8 VGPR): M = r + 8*(lane>=16), N = lane%16.
// ---------------------------------------------------------------------------
template <int ACT>
__global__ void wmma_gemm_f32(const float* __restrict__ A,
                              const float* __restrict__ W,
                              const float* __restrict__ bias,
                              float* __restrict__ C,
                              int Mm, int Nn, int Kk) {
  const int wavesPerBlock = blockDim.x >> 5;
  const int wave = blockIdx.x * wavesPerBlock + (threadIdx.x >> 5);
  const int lane = threadIdx.x & 31;
  const int stripsN = Nn >> 6;              // 64-column strips
  const int strips  = (Mm >> 4) * stripsN;
  if (wave >= strips) return;  // uniform per-wave: EXEC all-ones inside

  const int tm   = wave / stripsN;
  const int tn   = wave % stripsN;
  const int row0 = tm << 4;
  const int col0 = tn << 6;
  const int half = lane >> 4;    // 0: K pair {k,k+1}, 1: K pair {k+2,k+3}
  const int l16  = lane & 15;

  v8f acc0 = {}, acc1 = {}, acc2 = {}, acc3 = {};
  const float* ap = A + (size_t)(row0 + l16) * Kk + 2 * half;
  const float* bp = W + (size_t)(2 * half) * Nn + col0 + l16;

  for (int k = 0; k < Kk; k += 4) {
    if (k + 8 < Kk) {  // uniform condition; prefetch B two K-steps ahead
      __builtin_prefetch(bp + (size_t)8 * Nn, 0, 1);
      __builtin_prefetch(bp + (size_t)9 * Nn, 0, 1);
    }
    v2f a;
    a.x = ap[0];
    a.y = ap[1];
    v2f b0, b1, b2, b3;
    b0.x = bp[0];  b0.y = bp[Nn];
    b1.x = bp[16]; b1.y = bp[Nn + 16];
    b2.x = bp[32]; b2.y = bp[Nn + 32];
    b3.x = bp[48]; b3.y = bp[Nn + 48];
    acc0 = __builtin_amdgcn_wmma_f32_16x16x4_f32(false, a, false, b0, (short)0,
                                                 acc0, false, false);
    acc1 = __builtin_amdgcn_wmma_f32_16x16x4_f32(false, a, false, b1, (short)0,
                                                 acc1, false, false);
    acc2 = __builtin_amdgcn_wmma_f32_16x16x4_f32(false, a, false, b2, (short)0,
                                                 acc2, false, false);
    acc3 = __builtin_amdgcn_wmma_f32_16x16x4_f32(false, a, false, b3, (short)0,
                                                 acc3, false, false);
    ap += 4;
    bp += (size_t)4 * Nn;
  }

  // Epilogue: bias + activation, 4 tiles.
  v8f* accs[4] = {&acc0, &acc1, &acc2, &acc3};
#pragma unroll
  for (int t = 0; t < 4; ++t) {
    const int col = col0 + t * 16 + l16;
    const float be = bias[col];
    v8f acc = *accs[t];
#pragma unroll
    for (int r = 0; r < 8; ++r) {
      float v = acc[r] + be;
      if (ACT == 1) v = fmaxf(v, 0.0f);
      if (ACT == 2) v = fabsf(v);
      C[(size_t)(row0 + r + 8 * half) * Nn + col] = v;
    }
  }
}

// ---------------------------------------------------------------------------
// Coalition kernel: one block (256 thr = 8 waves) per batch element.
// numer[s,i,a] = exclPrefix_j(perm_act)[g_i][a], cnt = g_i;
// coal[i,a] = (1/(N*S)) * sum_s numer * g_i.
// inputs[bt,i,0:20] = coal, inputs[bt,i,20:40] = act[bt,i,:].
// Per-wave private accumulators -> deterministic reduction order.
// ---------------------------------------------------------------------------
__global__ void coalition_kernel(const float* __restrict__ actions, // (2048,16,20)
                                 const int* __restrict__ gc,        // (2048*32,16)
                                 float* __restrict__ inp) {         // (2048,16,40)
  __shared__ float act_s[NAG * NACT];        // 320
  __shared__ float coalw[8][NAG * NACT];     // per-wave accum
  __shared__ float pe_s[8][NAG * NACT];      // per-wave exclusive prefix

  const int bt   = blockIdx.x;
  const int tid  = threadIdx.x;
  const int w    = tid >> 5;
  const int lane = tid & 31;

  for (int i = tid; i < NAG * NACT; i += 256)
    act_s[i] = actions[(size_t)bt * NAG * NACT + i];
  for (int i = lane; i < NAG * NACT; i += 32) coalw[w][i] = 0.0f;
  __syncthreads();

  const float invNS = 1.0f / (float)(NAG * NS);

  for (int sq = 0; sq < 4; ++sq) {
    const int s = w + sq * 8;
    const int* grow = gc + ((size_t)bt * NS + s) * NAG;
    int g[NAG];
#pragma unroll
    for (int j = 0; j < NAG; ++j) g[j] = grow[j];

    if (lane < NACT) {
      const int a = lane;
      float run = 0.0f;
#pragma unroll
      for (int j = 0; j < NAG; ++j) {
        pe_s[w][j * NACT + a] = run;         // exclusive prefix at position j
        run += act_s[g[j] * NACT + a];
      }
#pragma unroll
      for (int i2 = 0; i2 < NAG; ++i2) {
        const int gi = g[i2];
        coalw[w][i2 * NACT + a] += pe_s[w][gi * NACT + a] * (float)gi * invNS;
      }
    }
  }
  __syncthreads();

  for (int idx = tid; idx < NAG * NACT; idx += 256) {
    float sum = 0.0f;
#pragma unroll
    for (int ww = 0; ww < 8; ++ww) sum += coalw[ww][idx];
    const int i = idx / NACT, a = idx % NACT;
    float* row = inp + (size_t)bt * NAG * 2 * NACT + (size_t)i * 2 * NACT;
    row[a]        = sum;        // coal
    row[NACT + a] = act_s[idx]; // act
  }
}

// ---------------------------------------------------------------------------
// Final fused kernel: one block (256 thr) per batch element.
// hidden[n,e] = elu(sum_k inputs[n,k]*w1[k,e] + b1[e])
// y[n] = sum_e hidden*w_final + (relu(st@Wv1+bv1)@Wv2 + bv2)
// w_est = |y|;  q_tot = sum_n w_est * agent_qs.
// out layout: [0,2048) q_tot, [2048, 34816) w_est.
// ---------------------------------------------------------------------------
__global__ void final_kernel(const float* __restrict__ inp,   // (2048,16,40)
                             const float* __restrict__ w1,    // (2048,40,64)
                             const float* __restrict__ b1v,   // (2048,64)
                             const float* __restrict__ wfin,  // (2048,64)
                             const float* __restrict__ vh,    // (2048,64) relu'd
                             const float* __restrict__ Wv2,   // (64)
                             const float* __restrict__ bv2,   // (1)
                             const float* __restrict__ aq,    // (2048,16)
                             float* __restrict__ out) {
  __shared__ float in_s[NAG * 2 * NACT];   // 640
  __shared__ float yred[NAG][NE];
  __shared__ float vsh[NE];
  __shared__ float v_s;
  __shared__ float warr[NAG];

  const int bt  = blockIdx.x;
  const int tid = threadIdx.x;
  const int e   = tid & 63;
  const int grp = tid >> 6;   // 4 groups of 4 agents

  for (int i = tid; i < NAG * 2 * NACT; i += 256)
    in_s[i] = inp[(size_t)bt * NAG * 2 * NACT + i];
  if (grp == 0) vsh[e] = vh[(size_t)bt * NE + e] * Wv2[e];
  __syncthreads();

  const float b1e = b1v[(size_t)bt * NE + e];
  const float wfe = wfin[(size_t)bt * NE + e];
  const float* w1p = w1 + (size_t)bt * 2 * NACT * NE;

#pragma unroll
  for (int ag = 0; ag < 4; ++ag) {
    const int agent = grp * 4 + ag;
    float acc = b1e;
#pragma unroll
    for (int k = 0; k < 2 * NACT; ++k)
      acc = fmaf(in_s[agent * 2 * NACT + k], w1p[(size_t)k * NE + e], acc);
    const float hid = acc > 0.0f ? acc : expm1f(acc);  // elu
    yred[agent][e] = hid * wfe;
  }
  if (tid == 0) {
    float vs = bv2[0];
    for (int i = 0; i < NE; ++i) vs += vsh[i];
    v_s = vs;
  }
  __syncthreads();

  if (tid < NAG) {
    float y = v_s;
    for (int i = 0; i < NE; ++i) y += yred[tid][i];
    const float we = fabsf(y);
    warr[tid] = we;
    out[BATCH + (size_t)bt * NAG + tid] = we;   // w_est
  }
  __syncthreads();

  if (tid == 0) {
    float q = 0.0f;
    for (int i = 0; i < NAG; ++i) q += warr[i] * aq[(size_t)bt * NAG + i];
    out[bt] = q;                                // q_tot
  }
}

// ---------------------------------------------------------------------------
extern "C" void kernel_launch(void* const* d_in, const int* in_sizes, int n_in,
                              void* d_out, int out_size, void* d_ws, size_t ws_size,
                              hipStream_t stream) {
  (void)in_sizes; (void)n_in; (void)out_size; (void)ws_size;

  const float* states   = (const float*)d_in[0];   // (32,64,128)
  const float* actions  = (const float*)d_in[1];   // (32,64,16,20)
  const float* agent_qs = (const float*)d_in[2];   // (32,64,16)
  /* d_in[3] = max_filter: unused by the reference */
  const int*   gc       = (const int*)d_in[4];     // (65536,16)
  const float* W1a = (const float*)d_in[5];
  const float* b1a = (const float*)d_in[6];
  const float* W1b = (const float*)d_in[7];
  const float* b1b = (const float*)d_in[8];
  const float* Wb  = (const float*)d_in[9];
  const float* bb  = (const float*)d_in[10];
  const float* Wfa = (const float*)d_in[11];
  const float* bfa = (const float*)d_in[12];
  const float* Wfb = (const float*)d_in[13];
  const float* bfb = (const float*)d_in[14];
  const float* Wv1 = (const float*)d_in[15];
  const float* bv1 = (const float*)d_in[16];
  const float* Wv2 = (const float*)d_in[17];
  const float* bv2 = (const float*)d_in[18];
  float* out = (float*)d_out;

  // Workspace layout (floats)
  float* ws   = (float*)d_ws;
  float* h1   = ws;                                  // 2048*256
  float* hf   = h1   + (size_t)BATCH * 256;          // 2048*256
  float* w1   = hf   + (size_t)BATCH * 256;          // 2048*2560
  float* b1vp = w1   + (size_t)BATCH * 2560;         // 2048*64
  float* wfin = b1vp + (size_t)BATCH * NE;           // 2048*64
  float* vhb  = wfin + (size_t)BATCH * NE;           // 2048*64
  float* inpb = vhb  + (size_t)BATCH * NE;           // 2048*640

  const dim3 blk(256);                               // 8 waves per block
  auto nblk = [](int strips) { return (unsigned)((strips + 7) / 8); };
  const int mt = BATCH / 16;                         // 128 M-tiles

  // Hypernetwork (per batch element, since st_rep rows repeat N=16x)
  wmma_gemm_f32<1><<<nblk(mt * (256 / 64)),  blk, 0, stream>>>(states, W1a, b1a, h1,   BATCH, 256,  128);
  wmma_gemm_f32<1><<<nblk(mt * (256 / 64)),  blk, 0, stream>>>(states, Wfa, bfa, hf,   BATCH, 256,  128);
  wmma_gemm_f32<2><<<nblk(mt * (2560 / 64)), blk, 0, stream>>>(h1,     W1b, b1b, w1,   BATCH, 2560, 256);
  wmma_gemm_f32<2><<<nblk(mt * (64 / 64)),   blk, 0, stream>>>(hf,     Wfb, bfb, wfin, BATCH, 64,   256);
  wmma_gemm_f32<0><<<nblk(mt * (64 / 64)),   blk, 0, stream>>>(states, Wb,  bb,  b1vp, BATCH, 64,   128);
  wmma_gemm_f32<1><<<nblk(mt * (64 / 64)),   blk, 0, stream>>>(states, Wv1, bv1, vhb,  BATCH, 64,   128);

  coalition_kernel<<<BATCH, blk, 0, stream>>>(actions, gc, inpb);
  final_kernel<<<BATCH, blk, 0, stream>>>(inpb, w1, b1vp, wfin, vhb, Wv2, bv2,
                                          agent_qs, out);
}